// VNNGP_75376676044813
// MI455X (gfx1250) — compile-verified
//
#include <hip/hip_runtime.h>
#include <hip/hip_bf16.h>

// ---------------------------------------------------------------------------
// VNNGP predict pipeline for MI455X (gfx1250, wave32, WMMA fp32 path,
// double-buffered async global->LDS staging for the big GEMM)
//
// Problem sizes: M=2048 inducing, K=32 neighbors, DIM=16, N=50000 test points.
// Outputs (concatenated flat): mean[50000], std[50000], mu[2048],
//                              Lu[2048*2048], L[2048*2048]
// ---------------------------------------------------------------------------

#define MM     2048
#define DIMD   16
#define KNN    32
#define NPTS   50000
#define JIT    1.0e-4f
#define CB     64        // Cholesky block size
#define NPT    16        // test points per block in predict kernel
#define TS     64        // Sigma block tile (rows x cols)
#define TK     64        // Sigma K chunk
#define LDP    68        // padded LDS row stride (floats): bank = row*4+k mod 64

typedef __attribute__((ext_vector_type(2))) float v2f;
typedef __attribute__((ext_vector_type(8))) float v8f;

__device__ __forceinline__ v8f wmma4(v2f a, v2f b, v8f c) {
  // D = A(16x4,f32) * B(4x16,f32) + C(16x16,f32)
  return __builtin_amdgcn_wmma_f32_16x16x4_f32(
      /*neg_a=*/false, a, /*neg_b=*/false, b,
      /*c_mod=*/(short)0, c, /*reuse_a=*/false, /*reuse_b=*/false);
}

__device__ __forceinline__ v8f vzero8() {
  v8f z = {0.0f,0.0f,0.0f,0.0f,0.0f,0.0f,0.0f,0.0f};
  return z;
}

// CDNA5 async global->LDS copy (ASYNCcnt-tracked); lds_byte_off is the
// wave-relative LDS byte address (low 32 bits of a __shared__ pointer).
__device__ __forceinline__ void async_b128(uint32_t lds_byte_off, const float* g) {
  asm volatile("global_load_async_to_lds_b128 %0, %1, off"
               :: "v"(lds_byte_off), "v"(g) : "memory");
}
// Async loads complete in order: waiting <=8 drains everything older than the
// 8 most recently issued copies (the in-flight next-chunk batch).
__device__ __forceinline__ void wait_async8() {
  asm volatile("s_wait_asynccnt 0x8" ::: "memory");
}
__device__ __forceinline__ void wait_async0() {
  asm volatile("s_wait_asynccnt 0x0" ::: "memory");
}

// ---------------------------------------------------------------------------
// prep: z2[i] = |Z_i|^2, copy mu to output
// ---------------------------------------------------------------------------
__global__ void k_prep_small(const float* __restrict__ Z,
                             const float* __restrict__ mu,
                             float* __restrict__ z2,
                             float* __restrict__ mu_out) {
  int i = blockIdx.x * blockDim.x + threadIdx.x;
  if (i < MM) {
    float s = 0.0f;
    for (int k = 0; k < DIMD; ++k) {
      float v = Z[i * DIMD + k];
      s += v * v;
    }
    z2[i] = s;
    mu_out[i] = mu[i];
  }
}

// Lu = tril(Lu_raw,-1) + diag(exp(diag(Lu_raw)))
__global__ void k_prep_lu(const float* __restrict__ raw,
                          float* __restrict__ Lu) {
  int i = blockIdx.x * blockDim.x + threadIdx.x;
  int r = i / MM, c = i % MM;
  float v = raw[i];
  Lu[i] = (c < r) ? v : ((c == r) ? __expf(v) : 0.0f);
}

// ---------------------------------------------------------------------------
// Kzz = exp(-0.5*max(z2_m + z2_n - 2 Z.Z^T, 0)); Lwork = Kzz + JIT*I
// One wave computes one 16x16 tile; dot via 4x v_wmma_f32_16x16x4_f32.
// ---------------------------------------------------------------------------
__global__ void k_kzz(const float* __restrict__ Z,
                      const float* __restrict__ z2,
                      float* __restrict__ Kzz,
                      float* __restrict__ Lwork) {
  int tm = blockIdx.x * 16, tn = blockIdx.y * 16;
  int lane = threadIdx.x & 31;
  int row = lane & 15;
  int kp  = (lane >> 4) * 2;
  v8f acc = vzero8();
  for (int kb = 0; kb < DIMD; kb += 4) {
    v2f a, b;
    a.x = Z[(tm + row) * DIMD + kb + kp];
    a.y = Z[(tm + row) * DIMD + kb + kp + 1];
    b.x = Z[(tn + row) * DIMD + kb + kp];
    b.y = Z[(tn + row) * DIMD + kb + kp + 1];
    acc = wmma4(a, b, acc);
  }
  float zn = z2[tn + row];
  for (int r = 0; r < 8; ++r) {
    int m = tm + r + 8 * (lane >> 4);
    int n = tn + row;
    float d2 = fmaxf(z2[m] + zn - 2.0f * acc[r], 0.0f);
    float kz = __expf(-0.5f * d2);
    Kzz[m * MM + n]   = kz;
    Lwork[m * MM + n] = kz + ((m == n) ? JIT : 0.0f);
  }
}

// ---------------------------------------------------------------------------
// Sigma = Lu * Lu^T  (2048^3 fp32 GEMM)
// Block = 256 threads (8 waves) computes a 64x64 tile.  A/B 64x64 chunks are
// staged into double-buffered LDS with global_load_async_to_lds_b128
// (ASYNCcnt); the next chunk is prefetched while the current one feeds
// 2 x 16 v_wmma_f32_16x16x4_f32 per wave (padded stride 68 keeps both the
// 16B async stores and b64 fragment reads bank-conflict-free).
// ---------------------------------------------------------------------------
__global__ void __launch_bounds__(256)
k_sigma(const float* __restrict__ Lu, float* __restrict__ Sigma) {
  __shared__ float As[2][TS * LDP];
  __shared__ float Bs[2][TS * LDP];

  int tid  = threadIdx.x;
  int lane = tid & 31;
  int wid  = tid >> 5;                 // 0..7
  int tm = blockIdx.x * TS, tn = blockIdx.y * TS;
  int wy = wid & 3;                    // row subtile 0..3
  int wx = wid >> 2;                   // col pair    0..1
  int row = lane & 15;
  int kp  = (lane >> 4) * 2;

  uint32_t aBase0 = (uint32_t)(uintptr_t)(void*)&As[0][0];
  uint32_t aBase1 = (uint32_t)(uintptr_t)(void*)&As[1][0];
  uint32_t bBase0 = (uint32_t)(uintptr_t)(void*)&Bs[0][0];
  uint32_t bBase1 = (uint32_t)(uintptr_t)(void*)&Bs[1][0];
  int lr = tid >> 2;                   // 0..63: row this thread stages
  int ls = tid & 3;                    // chunk group within the row

  const float* gA = Lu + (size_t)(tm + lr) * MM;
  const float* gB = Lu + (size_t)(tn + lr) * MM;

#define SIGMA_ISSUE(abase, bbase, kc_)                                   \
  do {                                                                   \
    for (int i_ = 0; i_ < 4; ++i_) {                                     \
      int col_ = (ls + i_ * 4) * 4;                                      \
      uint32_t loff_ = (uint32_t)(lr * LDP + col_) * 4u;                 \
      async_b128((abase) + loff_, gA + (kc_) + col_);                    \
      async_b128((bbase) + loff_, gB + (kc_) + col_);                    \
    }                                                                    \
  } while (0)

  SIGMA_ISSUE(aBase0, bBase0, 0);      // prologue: chunk 0 -> buffer 0

  v8f acc0 = vzero8(), acc1 = vzero8();
  for (int kc = 0; kc < MM; kc += TK) {
    int cur = (kc / TK) & 1;
    if (kc + TK < MM) {
      if (cur == 0) SIGMA_ISSUE(aBase1, bBase1, kc + TK);
      else          SIGMA_ISSUE(aBase0, bBase0, kc + TK);
      wait_async8();                   // drain current chunk, keep next in flight
    } else {
      wait_async0();                   // last chunk: drain everything
    }
    __syncthreads();

    const float* Ar  = &As[cur][(wy * 16 + row) * LDP];
    const float* Br0 = &Bs[cur][(wx * 32 + row) * LDP];
    const float* Br1 = &Bs[cur][(wx * 32 + 16 + row) * LDP];
    for (int kk = 0; kk < TK; kk += 4) {
      v2f a, b0, b1;
      a.x  = Ar[kk + kp];   a.y  = Ar[kk + kp + 1];
      b0.x = Br0[kk + kp];  b0.y = Br0[kk + kp + 1];
      b1.x = Br1[kk + kp];  b1.y = Br1[kk + kp + 1];
      acc0 = wmma4(a, b0, acc0);
      acc1 = wmma4(a, b1, acc1);
    }
    __syncthreads();                   // all waves done with buf before reuse
  }
#undef SIGMA_ISSUE

  for (int r = 0; r < 8; ++r) {
    int m  = tm + wy * 16 + r + 8 * (lane >> 4);
    int n0 = tn + wx * 32 + row;
    Sigma[(size_t)m * MM + n0]      = acc0[r];
    Sigma[(size_t)m * MM + n0 + 16] = acc1[r];
  }
}

// ---------------------------------------------------------------------------
// Blocked right-looking Cholesky on Lwork (in place), block size CB=64.
// ---------------------------------------------------------------------------
__global__ void k_chol_diag(float* __restrict__ A, int kb) {
  __shared__ float t[CB][CB + 1];
  int tid = threadIdx.x;  // 0..63
  for (int j = 0; j < CB; ++j) t[tid][j] = A[(size_t)(kb + tid) * MM + kb + j];
  __syncthreads();
  for (int k = 0; k < CB; ++k) {
    if (tid == k) t[k][k] = sqrtf(t[k][k]);
    __syncthreads();
    if (tid > k) t[tid][k] /= t[k][k];
    __syncthreads();
    if (tid > k)
      for (int j = k + 1; j <= tid; ++j) t[tid][j] -= t[tid][k] * t[j][k];
    __syncthreads();
  }
  for (int j = 0; j < CB; ++j)
    A[(size_t)(kb + tid) * MM + kb + j] = (j <= tid) ? t[tid][j] : 0.0f;
}

// panel TRSM: T = A_panel * L_diag^-T, one 64x64 tile per block (64 threads)
__global__ void k_chol_trsm(float* __restrict__ A, int kb) {
  __shared__ float Ld[CB][CB + 1];
  __shared__ float T[CB][CB + 1];
  int tid = threadIdx.x;
  int rb = kb + CB + blockIdx.x * CB;
  for (int j = 0; j < CB; ++j) {
    Ld[tid][j] = A[(size_t)(kb + tid) * MM + kb + j];
    T[tid][j]  = A[(size_t)(rb + tid) * MM + kb + j];
  }
  __syncthreads();
  for (int j = 0; j < CB; ++j) {
    float s = T[tid][j];
    for (int i = 0; i < j; ++i) s -= T[tid][i] * Ld[j][i];
    T[tid][j] = s / Ld[j][j];
  }
  for (int j = 0; j < CB; ++j)
    A[(size_t)(rb + tid) * MM + kb + j] = T[tid][j];
}

// trailing SYRK update via WMMA: C -= P P^T (only lower-tri tiles)
__global__ void k_chol_syrk(float* __restrict__ A, int kb) {
  int it = blockIdx.x, jt = blockIdx.y;
  if (jt > it) return;
  int tb = kb + CB;
  int tm = tb + it * 16, tn = tb + jt * 16;
  int lane = threadIdx.x & 31;
  int row = lane & 15;
  int kp  = (lane >> 4) * 2;
  v8f acc = vzero8();
  for (int c = 0; c < CB; c += 4) {
    v2f a, b;
    a.x = A[(size_t)(tm + row) * MM + kb + c + kp];
    a.y = A[(size_t)(tm + row) * MM + kb + c + kp + 1];
    b.x = A[(size_t)(tn + row) * MM + kb + c + kp];
    b.y = A[(size_t)(tn + row) * MM + kb + c + kp + 1];
    acc = wmma4(a, b, acc);
  }
  for (int r = 0; r < 8; ++r) {
    int m = tm + r + 8 * (lane >> 4);
    int n = tn + row;
    A[(size_t)m * MM + n] -= acc[r];
  }
}

__global__ void k_extract_L(const float* __restrict__ Lwork,
                            float* __restrict__ Lout) {
  int i = blockIdx.x * blockDim.x + threadIdx.x;
  int r = i / MM, c = i % MM;
  Lout[i] = (c <= r) ? Lwork[i] : 0.0f;
}

// ---------------------------------------------------------------------------
// Fused per-point kernel: WMMA distances -> top-32 -> 32x32 SPD solve ->
// mean / std.  Block = 128 threads (4 waves), NPT=16 test points per block.
// LDS: d2 slab 16x2048 fp32 = 128KB (reused as solve scratch), small buffers.
// ---------------------------------------------------------------------------
__global__ void __launch_bounds__(128)
k_predict(const float* __restrict__ X, const float* __restrict__ Z,
          const float* __restrict__ z2, const float* __restrict__ Kzz,
          const float* __restrict__ Sigma, const float* __restrict__ mu,
          float* __restrict__ mean_out, float* __restrict__ std_out) {
  __shared__ float d2s[NPT][MM];       // 128 KB; reused as solve scratch
  __shared__ float x2s[NPT];
  __shared__ int   idxs[NPT][KNN];
  __shared__ float dsel[NPT][KNN];

  int tid  = threadIdx.x;
  int lane = tid & 31;
  int wid  = tid >> 5;                 // 0..3
  int pbase = blockIdx.x * NPT;

  // |x|^2 for the block's 16 points
  if (tid < NPT) {
    float s = 0.0f;
    for (int k = 0; k < DIMD; ++k) {
      float v = X[(size_t)(pbase + tid) * DIMD + k];
      s += v * v;
    }
    x2s[tid] = s;
  }
  __syncthreads();

  // ---- stage 1: d2(point, all 2048 Z) via WMMA (16x16 tiles, K=16) ----
  {
    int row = lane & 15;
    int kp  = (lane >> 4) * 2;
    v2f afrag[4];
    for (int c = 0; c < 4; ++c) {
      afrag[c].x = X[(size_t)(pbase + row) * DIMD + c * 4 + kp];
      afrag[c].y = X[(size_t)(pbase + row) * DIMD + c * 4 + kp + 1];
    }
    float xm[8];
    for (int r = 0; r < 8; ++r) xm[r] = x2s[r + 8 * (lane >> 4)];
    for (int t = wid; t < MM / 16; t += 4) {
      int zb = t * 16;
      __builtin_prefetch(Z + (size_t)(zb + 64) * DIMD, 0, 1);
      v8f acc = vzero8();
      for (int c = 0; c < 4; ++c) {
        v2f b;
        b.x = Z[(size_t)(zb + row) * DIMD + c * 4 + kp];
        b.y = Z[(size_t)(zb + row) * DIMD + c * 4 + kp + 1];
        acc = wmma4(afrag[c], b, acc);
      }
      float zn = z2[zb + row];
      for (int r = 0; r < 8; ++r) {
        int m = r + 8 * (lane >> 4);
        d2s[m][zb + row] = fmaxf(xm[r] + zn - 2.0f * acc[r], 0.0f);
      }
    }
  }
  __syncthreads();

  // ---- stage 2: top-32 per point (wave handles 4 points), stable order ----
  for (int pp = 0; pp < 4; ++pp) {
    int p = wid * 4 + pp;
    for (int s = 0; s < KNN; ++s) {
      float bv = 3.4e38f;
      int   bi = MM;
      for (int i = 0; i < MM / 32; ++i) {
        int e = i * 32 + lane;         // lanes hit distinct LDS banks
        float v = d2s[p][e];
        if (v < bv || (v == bv && e < bi)) { bv = v; bi = e; }
      }
      for (int off = 16; off > 0; off >>= 1) {
        float ov = __shfl_xor(bv, off, 32);
        int   oi = __shfl_xor(bi, off, 32);
        if (ov < bv || (ov == bv && oi < bi)) { bv = ov; bi = oi; }
      }
      if (lane == 0) {
        idxs[p][s] = bi;
        dsel[p][s] = bv;
        d2s[p][bi] = 3.4e38f;          // knock out for next pass
      }
      __syncthreads();                 // uniform trip counts across all waves
    }
  }
  __syncthreads();

  // ---- stage 3: per point 32x32 SPD solve + quadratic form ----
  float* flat = &d2s[0][0];            // reuse 128KB slab as scratch
  float* Am = flat + wid * 2176;       // 32x33 = 1056, padded to 1088
  float* Bm = Am + 1088;
  float* yv = flat + 4 * 2176 + wid * 64;   // y[32] then w[32]
  float* wv = yv + 32;

  for (int pp = 0; pp < 4; ++pp) {
    int p = wid * 4 + pp;
    int myidx = idxs[p][lane];
    // gather A = kzz_nn + JIT*I (row `lane`), B = kzz_nn - Sigma_nn
    for (int k = 0; k < KNN; ++k) {
      int ik = idxs[p][k];
      float kz = Kzz[(size_t)myidx * MM + ik];
      float sg = Sigma[(size_t)myidx * MM + ik];
      Am[lane * 33 + k] = kz + ((k == lane) ? JIT : 0.0f);
      Bm[lane * 33 + k] = kz - sg;
    }
    __syncthreads();

    // Cholesky A = L L^T, in place (lower)
    for (int k = 0; k < KNN; ++k) {
      if (lane == k) Am[k * 33 + k] = sqrtf(Am[k * 33 + k]);
      __syncthreads();
      if (lane > k) Am[lane * 33 + k] /= Am[k * 33 + k];
      __syncthreads();
      if (lane > k)
        for (int j = k + 1; j <= lane; ++j)
          Am[lane * 33 + j] -= Am[lane * 33 + k] * Am[j * 33 + k];
      __syncthreads();
    }

    // forward solve L y = kzx  (kzx = exp(-0.5 * d2_selected))
    for (int k = 0; k < KNN; ++k) {
      float c = (lane < k) ? Am[k * 33 + lane] * yv[lane] : 0.0f;
      for (int off = 16; off > 0; off >>= 1) c += __shfl_xor(c, off, 32);
      if (lane == 0) {
        float kx = __expf(-0.5f * dsel[p][k]);
        yv[k] = (kx - c) / Am[k * 33 + k];
      }
      __syncthreads();
    }
    // backward solve L^T w = y
    for (int k = KNN - 1; k >= 0; --k) {
      float c = (lane > k) ? Am[lane * 33 + k] * wv[lane] : 0.0f;
      for (int off = 16; off > 0; off >>= 1) c += __shfl_xor(c, off, 32);
      if (lane == 0) wv[k] = (yv[k] - c) / Am[k * 33 + k];
      __syncthreads();
    }

    // mean = w . mu[idx];  cov = 1 - w^T B w
    float wl = wv[lane];
    float mv = wl * mu[myidx];
    float qs = 0.0f;
    for (int k = 0; k < KNN; ++k) qs += wv[k] * Bm[k * 33 + lane];
    float qv = wl * qs;
    for (int off = 16; off > 0; off >>= 1) {
      mv += __shfl_xor(mv, off, 32);
      qv += __shfl_xor(qv, off, 32);
    }
    if (lane == 0) {
      mean_out[pbase + p] = mv;
      std_out[pbase + p]  = sqrtf(fmaxf(1.0f - qv, 0.05f));
    }
    __syncthreads();
  }
}

// ---------------------------------------------------------------------------
// host launcher
// ---------------------------------------------------------------------------
extern "C" void kernel_launch(void* const* d_in, const int* in_sizes, int n_in,
                              void* d_out, int out_size, void* d_ws, size_t ws_size,
                              hipStream_t stream) {
  (void)in_sizes; (void)n_in; (void)out_size; (void)ws_size;
  const float* X      = (const float*)d_in[0];   // 50000 x 16
  const float* Z      = (const float*)d_in[1];   // 2048 x 16
  const float* Lu_raw = (const float*)d_in[2];   // 2048 x 2048
  const float* mu     = (const float*)d_in[3];   // 2048

  float* out    = (float*)d_out;
  float* mean_o = out;
  float* std_o  = out + NPTS;
  float* mu_o   = out + 2 * NPTS;
  float* Lu_o   = out + 2 * NPTS + MM;
  float* L_o    = Lu_o + (size_t)MM * MM;

  float* ws    = (float*)d_ws;
  float* Kzz   = ws;
  float* Lwork = ws + (size_t)MM * MM;
  float* Sigma = ws + 2 * (size_t)MM * MM;
  float* z2    = ws + 3 * (size_t)MM * MM;

  k_prep_small<<<(MM + 255) / 256, 256, 0, stream>>>(Z, mu, z2, mu_o);
  k_prep_lu<<<(MM * MM) / 256, 256, 0, stream>>>(Lu_raw, Lu_o);
  k_kzz<<<dim3(MM / 16, MM / 16), 32, 0, stream>>>(Z, z2, Kzz, Lwork);
  k_sigma<<<dim3(MM / TS, MM / TS), 256, 0, stream>>>(Lu_o, Sigma);

  for (int s = 0; s < MM / CB; ++s) {
    int kb = s * CB;
    k_chol_diag<<<1, CB, 0, stream>>>(Lwork, kb);
    int rem = MM - kb - CB;
    if (rem > 0) {
      k_chol_trsm<<<rem / CB, CB, 0, stream>>>(Lwork, kb);
      int nt = rem / 16;
      k_chol_syrk<<<dim3(nt, nt), 32, 0, stream>>>(Lwork, kb);
    }
  }
  k_extract_L<<<(MM * MM) / 256, 256, 0, stream>>>(Lwork, L_o);

  k_predict<<<NPTS / NPT, 128, 0, stream>>>(X, Z, z2, Kzz, Sigma, mu,
                                            mean_o, std_o);
}